// GraphConvClassifier_25640954757642
// MI455X (gfx1250) — compile-verified
//
#include <hip/hip_runtime.h>
#include <hip/hip_bf16.h>
#include <math.h>

#define N_NODES   100000
#define N_EDGES   1600000
#define IN_DIM    64
#define HID       128
#define N_GRAPHS  64
#define N_CLASSES 2
#define EPSV      1e-5f

typedef __attribute__((ext_vector_type(2))) float v2f;
typedef __attribute__((ext_vector_type(8))) float v8f;

// ---------------------------------------------------------------------------
// Utility fills
// ---------------------------------------------------------------------------
__global__ void k_fill(float* __restrict__ p, int n, float v) {
    int i = blockIdx.x * blockDim.x + threadIdx.x;
    if (i < n) p[i] = v;
}

// deg[dst] += 1 for each edge (self-loop handled by init deg=1)
__global__ void k_edge_deg(const int* __restrict__ dst, float* __restrict__ deg, int nE) {
    int i = blockIdx.x * blockDim.x + threadIdx.x;
    if (i < nE) atomicAdd(&deg[dst[i]], 1.0f);
}

// cnt[batch[i]] += 1
__global__ void k_node_cnt(const int* __restrict__ batch, float* __restrict__ cnt, int n) {
    int i = blockIdx.x * blockDim.x + threadIdx.x;
    if (i < n) atomicAdd(&cnt[batch[i]], 1.0f);
}

__global__ void k_rsqrt(const float* __restrict__ deg, float* __restrict__ dis, int n) {
    int i = blockIdx.x * blockDim.x + threadIdx.x;
    if (i < n) dis[i] = rsqrtf(deg[i]);
}

// ---------------------------------------------------------------------------
// f32 WMMA GEMM with fused self-loop + bias epilogue:
//   XW[N x 128]  = H[N x K] @ W[K x 128]                  (edge-gather source)
//   Agg[N x 128] = XW * dis[row]^2 + bias[col]            (scatter accumulator init)
// Block = 256 threads = 8 waves; block covers 16 rows x 128 cols.
// Wave w computes the 16x16 tile at columns [16w, 16w+16).
// A tile (16 x K) staged in LDS, shared by all 8 waves.
// ---------------------------------------------------------------------------
template<int K>
__global__ void k_gemm_wmma(const float* __restrict__ H, const float* __restrict__ W,
                            const float* __restrict__ dis, const float* __restrict__ bias,
                            float* __restrict__ XW, float* __restrict__ Agg) {
    __shared__ float As[16 * K];
    const int r0 = blockIdx.x * 16;

    // cooperative load of the 16 x K A-tile
    for (int idx = threadIdx.x; idx < 16 * K; idx += 256) {
        int row = idx / K, col = idx - row * K;
        As[idx] = H[(size_t)(r0 + row) * K + col];
    }
    __syncthreads();

    const int wave = threadIdx.x >> 5;   // 0..7 -> column tile
    const int lane = threadIdx.x & 31;
    const int half = lane >> 4;          // lane-half selects K sub-pair / row-half
    const int m    = lane & 15;
    const int c0   = wave * 16;

    v8f acc = {};
    for (int k = 0; k < K; k += 4) {
        v2f a, b;
        // A 16x4 f32 layout: lanes 0-15: (K=k+0, K=k+1); lanes 16-31: (K=k+2, K=k+3)
        a.x = As[m * K + k + 2 * half + 0];
        a.y = As[m * K + k + 2 * half + 1];
        // B 4x16 f32 layout: lanes hold N=m; lane-half selects K rows {k,k+1} vs {k+2,k+3}
        b.x = W[(size_t)(k + 2 * half + 0) * HID + c0 + m];
        b.y = W[(size_t)(k + 2 * half + 1) * HID + c0 + m];
        acc = __builtin_amdgcn_wmma_f32_16x16x4_f32(
                  /*neg_a=*/false, a, /*neg_b=*/false, b,
                  /*c_mod=*/(short)0, acc, /*reuse_a=*/false, /*reuse_b=*/false);
    }

    // C/D layout: VGPR v holds row r0+v (lanes 0-15) and r0+v+8 (lanes 16-31), col c0+m
    const float bc = bias[c0 + m];
#pragma unroll
    for (int v = 0; v < 8; ++v) {
        const int row = r0 + v + 8 * half;
        const float d = dis[row];
        const size_t off = (size_t)row * HID + c0 + m;
        XW[off]  = acc[v];
        Agg[off] = acc[v] * (d * d) + bc;
    }
}

// ---------------------------------------------------------------------------
// per-edge scatter: agg[dst] += XW[src] * dis[src]*dis[dst]
// 8 edges per 256-thread block; 32 lanes/edge, each lane gathers a float4
// (global_load_b128) and issues 4 global_atomic_add_f32.
// ---------------------------------------------------------------------------
__global__ void k_edge_scatter(const int* __restrict__ src, const int* __restrict__ dst,
                               const float* __restrict__ dis, const float* __restrict__ XW,
                               float* __restrict__ agg, int nE) {
    int e = blockIdx.x * 8 + (threadIdx.x >> 5);
    if (e >= nE) return;
    int lane = threadIdx.x & 31;
    int s = src[e], d = dst[e];
    float w = dis[s] * dis[d];
    const float4 v = ((const float4*)(XW + (size_t)s * HID))[lane];
    float* ap = agg + (size_t)d * HID + lane * 4;
    atomicAdd(ap + 0, v.x * w);
    atomicAdd(ap + 1, v.y * w);
    atomicAdd(ap + 2, v.z * w);
    atomicAdd(ap + 3, v.w * w);
}

// ---------------------------------------------------------------------------
// Segment sums (sum, sumsq) per (graph, channel). batch is sorted, so each
// thread (1 per channel) run-length-compresses before touching global atomics.
// Block covers NODES_PER_BLOCK consecutive nodes.
// ---------------------------------------------------------------------------
#define NODES_PER_BLOCK 64
__global__ void k_stats(const float* __restrict__ h, const int* __restrict__ batch,
                        float* __restrict__ sum, float* __restrict__ sumsq, int n) {
    int c = threadIdx.x;                      // 0..127
    int base = blockIdx.x * NODES_PER_BLOCK;
    float s = 0.f, q = 0.f;
    int cur = -1;
    for (int i = 0; i < NODES_PER_BLOCK; ++i) {
        int node = base + i;
        if (node >= n) break;                 // uniform across the block
        int g = batch[node];                  // uniform -> scalar load
        if (g != cur) {
            if (cur >= 0) {
                atomicAdd(&sum[cur * HID + c], s);
                atomicAdd(&sumsq[cur * HID + c], q);
            }
            cur = g; s = 0.f; q = 0.f;
        }
        float v = h[(size_t)node * HID + c];
        s += v; q += v * v;
    }
    if (cur >= 0) {
        atomicAdd(&sum[cur * HID + c], s);
        atomicAdd(&sumsq[cur * HID + c], q);
    }
}

// mean = sum/cnt ; inv = rsqrt(var + eps)   (in-place over sum/sumsq)
__global__ void k_meaninv(float* __restrict__ sum, float* __restrict__ sumsq,
                          const float* __restrict__ cnt) {
    int idx = blockIdx.x * blockDim.x + threadIdx.x;      // 64*128
    if (idx >= N_GRAPHS * HID) return;
    int g = idx >> 7;
    float c = fmaxf(cnt[g], 1.0f);
    float m = sum[idx] / c;
    float v = sumsq[idx] / c - m * m;
    sum[idx]   = m;
    sumsq[idx] = rsqrtf(v + EPSV);
}

// h_out = relu((h_in - mean[g][c]) * inv[g][c]) — float4 vectorized
__global__ void k_norm_relu(const float* __restrict__ hin, const int* __restrict__ batch,
                            const float* __restrict__ mean, const float* __restrict__ inv,
                            float* __restrict__ hout) {
    int idx = blockIdx.x * blockDim.x + threadIdx.x;      // N_NODES * (HID/4)
    if (idx >= N_NODES * (HID / 4)) return;
    int i = idx >> 5;                 // node
    int c4 = idx & 31;                // float4 channel group
    int g = batch[i];
    float4 h  = ((const float4*)hin)[idx];
    float4 mu = ((const float4*)(mean + g * HID))[c4];
    float4 iv = ((const float4*)(inv  + g * HID))[c4];
    float4 r;
    r.x = fmaxf((h.x - mu.x) * iv.x, 0.0f);
    r.y = fmaxf((h.y - mu.y) * iv.y, 0.0f);
    r.z = fmaxf((h.z - mu.z) * iv.z, 0.0f);
    r.w = fmaxf((h.w - mu.w) * iv.w, 0.0f);
    ((float4*)hout)[idx] = r;
}

// ---------------------------------------------------------------------------
// Head: pooled = sum3/cnt ; BatchNorm over graphs (per channel) ; z @ fcW + fcb
// Single block, 128 threads (one per channel), z staged in LDS.
// ---------------------------------------------------------------------------
__global__ void k_head(const float* __restrict__ sum3, const float* __restrict__ cnt,
                       const float* __restrict__ gamma, const float* __restrict__ beta,
                       const float* __restrict__ fcW, const float* __restrict__ fcb,
                       float* __restrict__ out) {
    __shared__ float z[N_GRAPHS * HID];
    int c = threadIdx.x;                      // channel
    float m = 0.f;
    for (int g = 0; g < N_GRAPHS; ++g)
        m += sum3[g * HID + c] / fmaxf(cnt[g], 1.0f);
    m *= (1.0f / N_GRAPHS);
    float v = 0.f;
    for (int g = 0; g < N_GRAPHS; ++g) {
        float p = sum3[g * HID + c] / fmaxf(cnt[g], 1.0f);
        float d = p - m;
        v += d * d;
    }
    v *= (1.0f / N_GRAPHS);
    float inv = rsqrtf(v + EPSV);
    float ga = gamma[c], be = beta[c];
    for (int g = 0; g < N_GRAPHS; ++g) {
        float p = sum3[g * HID + c] / fmaxf(cnt[g], 1.0f);
        z[g * HID + c] = (p - m) * inv * ga + be;
    }
    __syncthreads();
    // 128 outputs (64 graphs x 2 classes): thread t -> (g = t/2, k = t%2)
    int g = c >> 1, k = c & 1;
    float acc = fcb[k];
    for (int cc = 0; cc < HID; ++cc)
        acc += z[g * HID + cc] * fcW[cc * N_CLASSES + k];
    out[g * N_CLASSES + k] = acc;
}

// ---------------------------------------------------------------------------
extern "C" void kernel_launch(void* const* d_in, const int* in_sizes, int n_in,
                              void* d_out, int out_size, void* d_ws, size_t ws_size,
                              hipStream_t stream) {
    const float* x     = (const float*)d_in[0];
    const int*   eidx  = (const int*)  d_in[1];   // [2, N_EDGES]
    const int*   batch = (const int*)  d_in[2];
    const float* W1    = (const float*)d_in[3];
    const float* b1    = (const float*)d_in[4];
    const float* W2    = (const float*)d_in[5];
    const float* b2    = (const float*)d_in[6];
    const float* gamma = (const float*)d_in[7];
    const float* beta  = (const float*)d_in[8];
    const float* fcW   = (const float*)d_in[9];
    const float* fcb   = (const float*)d_in[10];
    float* out = (float*)d_out;

    const int* src = eidx;
    const int* dst = eidx + N_EDGES;

    // workspace layout (floats)
    float* ws    = (float*)d_ws;
    float* deg   = ws;                         size_t o = N_NODES;
    float* dis   = ws + o;                     o += N_NODES;
    float* bufA  = ws + o;                     o += (size_t)N_NODES * HID;
    float* bufB  = ws + o;                     o += (size_t)N_NODES * HID;
    float* sum   = ws + o;                     o += N_GRAPHS * HID;
    float* sumsq = ws + o;                     o += N_GRAPHS * HID;
    float* cnt   = ws + o;                     o += N_GRAPHS;
    (void)ws_size; (void)in_sizes; (void)n_in; (void)out_size;

    const int T = 256;
    const int nodeBlk  = (N_NODES + T - 1) / T;
    const int vecBlk   = (N_NODES * (HID / 4) + T - 1) / T;
    const int edgeBlk  = (N_EDGES + T - 1) / T;
    const int scatBlk  = (N_EDGES + 7) / 8;
    const int gemmBlk  = N_NODES / 16;              // 6250
    const int statBlk  = (N_NODES + NODES_PER_BLOCK - 1) / NODES_PER_BLOCK;
    const int gcBlk    = (N_GRAPHS * HID + T - 1) / T;

    // ---- degrees / counts -------------------------------------------------
    k_fill<<<nodeBlk, T, 0, stream>>>(deg, N_NODES, 1.0f);           // self-loops
    k_fill<<<1, T, 0, stream>>>(cnt, N_GRAPHS, 0.0f);
    k_fill<<<gcBlk, T, 0, stream>>>(sum,   N_GRAPHS * HID, 0.0f);
    k_fill<<<gcBlk, T, 0, stream>>>(sumsq, N_GRAPHS * HID, 0.0f);
    k_edge_deg<<<edgeBlk, T, 0, stream>>>(dst, deg, N_EDGES);
    k_node_cnt<<<nodeBlk, T, 0, stream>>>(batch, cnt, N_NODES);
    k_rsqrt<<<nodeBlk, T, 0, stream>>>(deg, dis, N_NODES);

    // ---- layer 1: fused GEMM+self-loop+bias, scatter, inst-norm, relu -----
    k_gemm_wmma<IN_DIM><<<gemmBlk, T, 0, stream>>>(x, W1, dis, b1, bufA, bufB);
    k_edge_scatter<<<scatBlk, T, 0, stream>>>(src, dst, dis, bufA, bufB, N_EDGES);
    k_stats<<<statBlk, HID, 0, stream>>>(bufB, batch, sum, sumsq, N_NODES);
    k_meaninv<<<gcBlk, T, 0, stream>>>(sum, sumsq, cnt);
    k_norm_relu<<<vecBlk, T, 0, stream>>>(bufB, batch, sum, sumsq, bufA);   // h1 -> bufA

    // ---- layer 2 ----------------------------------------------------------
    k_fill<<<gcBlk, T, 0, stream>>>(sum,   N_GRAPHS * HID, 0.0f);
    k_fill<<<gcBlk, T, 0, stream>>>(sumsq, N_GRAPHS * HID, 0.0f);
    k_gemm_wmma<HID><<<gemmBlk, T, 0, stream>>>(bufA, W2, dis, b2, bufB, bufA); // XW2->bufB, agg2->bufA
    k_edge_scatter<<<scatBlk, T, 0, stream>>>(src, dst, dis, bufB, bufA, N_EDGES);
    k_stats<<<statBlk, HID, 0, stream>>>(bufA, batch, sum, sumsq, N_NODES);
    k_meaninv<<<gcBlk, T, 0, stream>>>(sum, sumsq, cnt);
    k_norm_relu<<<vecBlk, T, 0, stream>>>(bufA, batch, sum, sumsq, bufB);   // h2 -> bufB

    // ---- pool + batch-norm + FC ------------------------------------------
    k_fill<<<gcBlk, T, 0, stream>>>(sum,   N_GRAPHS * HID, 0.0f);
    k_fill<<<gcBlk, T, 0, stream>>>(sumsq, N_GRAPHS * HID, 0.0f);
    k_stats<<<statBlk, HID, 0, stream>>>(bufB, batch, sum, sumsq, N_NODES);
    k_head<<<1, HID, 0, stream>>>(sum, cnt, gamma, beta, fcW, fcb, out);
}